// BertFusionAttCRFTokenClassModel_74311524156054
// MI455X (gfx1250) — compile-verified
//
#include <hip/hip_runtime.h>
#include <math.h>

// CRF loss for B=2048, T=512, L=23.
// Forward algorithm rewritten in exp-space: q = E*p (E = exp(transitions), constant),
// q *= exp(feat_t); renormalize by per-column max every 4 steps; running log-scale S.
// 23x23 (padded 32x32) matvec batched over 16 sequences -> v_wmma_f32_16x16x4_f32.
// Emission weights are SOFTWARE-PIPELINED one step ahead: loads + v_exp for step t+1
// are issued during step t, so their latency is hidden behind a full scan step.

#define NUM_LABELS 23
#define START_LABEL 21
#define TDIM 512
#define BDIM 2048
#define LPAD 32

typedef __attribute__((ext_vector_type(2))) float v2f;
typedef __attribute__((ext_vector_type(8))) float v8f;

__global__ __launch_bounds__(128) void crf_forward_kernel(
    const float* __restrict__ feats,
    const float* __restrict__ transitions,
    float* __restrict__ fwd_out)
{
    __shared__ __align__(16) float eLds[LPAD * LPAD];     // exp(transitions), zero-padded
    __shared__ __align__(16) float qLds[4][LPAD * 16];    // per-wave p, pair-interleaved

    const int tid = threadIdx.x;

    // Build E = exp(transitions) padded to 32x32 (pad entries exactly 0).
    #pragma unroll
    for (int idx = tid; idx < LPAD * LPAD; idx += 128) {
        int r = idx >> 5, c = idx & 31;
        float v = 0.0f;
        if (r < NUM_LABELS && c < NUM_LABELS) v = __expf(transitions[r * NUM_LABELS + c]);
        eLds[idx] = v;
    }
    __syncthreads();

    const int wave = tid >> 5;
    const int lane = tid & 31;
    const int half = lane >> 4;   // which 16-lane half
    const int l16  = lane & 15;
    const int bcol = (blockIdx.x * 4 + wave) * 16 + l16;  // this lane's batch column

    float* q = qLds[wave];

    // A-operands: E in WMMA A-layout (16x4 f32 tiles). 2 M-tiles x 8 K-tiles, loaded once.
    v2f A[2][8];
    #pragma unroll
    for (int m = 0; m < 2; ++m) {
        #pragma unroll
        for (int k = 0; k < 8; ++k) {
            const int row = m * 16 + l16;
            const int kb  = k * 4 + half * 2;
            v2f a;
            a.x = eLds[row * LPAD + kb];
            a.y = eLds[row * LPAD + kb + 1];
            A[m][k] = a;
        }
    }

    // Initial p0 (= exp(alpha0)): one-hot at START_LABEL. Pair-interleaved LDS layout:
    // float index of (row,col) = ((row>>1)*16 + col)*2 + (row&1)  -> rows (2r,2r+1) adjacent 8B.
    #pragma unroll
    for (int m = 0; m < 2; ++m) {
        #pragma unroll
        for (int i = 0; i < 4; ++i) {
            const int row0 = m * 16 + half * 8 + 2 * i;
            v2f v;
            v.x = (row0     == START_LABEL) ? 1.0f : 0.0f;
            v.y = (row0 + 1 == START_LABEL) ? 1.0f : 0.0f;
            *(v2f*)&q[((row0 >> 1) * 16 + l16) * 2] = v;
        }
    }

    float S = 0.0f;  // running log-scale of this lane's column
    float dv[16];    // this lane's 16 C/D elements (rows m*16+half*8+r, col l16)
    const float* fb = feats + (size_t)bcol * (TDIM * NUM_LABELS);

    // ---- Pipeline prologue: emission weights for t = 1 ----
    float ew[16];
    {
        const float* ft = fb + 1 * NUM_LABELS;
        #pragma unroll
        for (int m = 0; m < 2; ++m) {
            #pragma unroll
            for (int r = 0; r < 8; ++r) {
                const int row = m * 16 + half * 8 + r;
                const int rc  = row < NUM_LABELS ? row : (NUM_LABELS - 1);
                float w = __expf(ft[rc]);
                ew[m * 8 + r] = (row < NUM_LABELS) ? w : 0.0f;
            }
        }
    }

    for (int t = 1; t < TDIM; ++t) {
        // ---- Issue NEXT step's emission loads first (consumed only after the backedge) ----
        const int tn = (t + 1 < TDIM) ? (t + 1) : (TDIM - 1);   // clamp; last iter redundant
        const float* ftn = fb + tn * NUM_LABELS;
        __builtin_prefetch(ftn + NUM_LABELS, 0, 3);             // t+2 row head
        __builtin_prefetch(ftn + 2 * NUM_LABELS - 1, 0, 3);     // t+2 row tail
        float fwn[16];
        #pragma unroll
        for (int m = 0; m < 2; ++m) {
            #pragma unroll
            for (int r = 0; r < 8; ++r) {
                const int row = m * 16 + half * 8 + r;
                const int rc  = row < NUM_LABELS ? row : (NUM_LABELS - 1);
                fwn[m * 8 + r] = ftn[rc];
            }
        }

        // B-operands: previous p from LDS. Tile k needs rows {4k+2*half, +1} at col l16.
        v2f Bv[8];
        #pragma unroll
        for (int k = 0; k < 8; ++k) {
            const int pair = k * 2 + half;
            Bv[k] = *(const v2f*)&q[(pair * 16 + l16) * 2];
        }

        // q = E * p : 16 WMMAs, 4 independent accumulation chains for ILP.
        v8f acc0a = {}, acc0b = {}, acc1a = {}, acc1b = {};
        #pragma unroll
        for (int k = 0; k < 4; ++k) {
            acc0a = __builtin_amdgcn_wmma_f32_16x16x4_f32(false, A[0][k],     false, Bv[k],     (short)0, acc0a, false, false);
            acc0b = __builtin_amdgcn_wmma_f32_16x16x4_f32(false, A[0][k + 4], false, Bv[k + 4], (short)0, acc0b, false, false);
            acc1a = __builtin_amdgcn_wmma_f32_16x16x4_f32(false, A[1][k],     false, Bv[k],     (short)0, acc1a, false, false);
            acc1b = __builtin_amdgcn_wmma_f32_16x16x4_f32(false, A[1][k + 4], false, Bv[k + 4], (short)0, acc1b, false, false);
        }
        v8f d0 = acc0a + acc0b;
        v8f d1 = acc1a + acc1b;

        // q *= exp(feat_t[row]) -- weights already in registers from previous iteration.
        #pragma unroll
        for (int r = 0; r < 8; ++r) {
            dv[r]     = d0[r] * ew[r];
            dv[8 + r] = d1[r] * ew[8 + r];
        }

        // Renormalize every 4 steps (wave-uniform branch; EXEC stays full for WMMAs).
        if ((t & 3) == 0) {
            float mx = 0.0f;
            #pragma unroll
            for (int i = 0; i < 16; ++i) mx = fmaxf(mx, dv[i]);
            mx = fmaxf(mx, __shfl_xor(mx, 16, 32));
            const float inv = 1.0f / mx;
            S += __logf(mx);
            #pragma unroll
            for (int i = 0; i < 16; ++i) dv[i] *= inv;
        }

        // Store p back to LDS (pair-interleaved) for next step's B-operands.
        #pragma unroll
        for (int m = 0; m < 2; ++m) {
            #pragma unroll
            for (int i = 0; i < 4; ++i) {
                const int row0 = m * 16 + half * 8 + 2 * i;
                v2f v;
                v.x = dv[m * 8 + 2 * i];
                v.y = dv[m * 8 + 2 * i + 1];
                *(v2f*)&q[((row0 >> 1) * 16 + l16) * 2] = v;
            }
        }

        // ---- Pipeline epilogue: exp weights for step t+1 (v_exp overlaps everything above) ----
        #pragma unroll
        for (int m = 0; m < 2; ++m) {
            #pragma unroll
            for (int r = 0; r < 8; ++r) {
                const int row = m * 16 + half * 8 + r;
                float w = __expf(fwn[m * 8 + r]);
                ew[m * 8 + r] = (row < NUM_LABELS) ? w : 0.0f;  // pad rows stay exactly 0
            }
        }
    }

    // forward_score = log(sum_rows p) + S (last few steps may be un-renormalized; safe range).
    float s = 0.0f;
    #pragma unroll
    for (int i = 0; i < 16; ++i) s += dv[i];
    s += __shfl_xor(s, 16, 32);
    if (half == 0) fwd_out[bcol] = __logf(s) + S;
}

__global__ __launch_bounds__(256) void crf_gold_kernel(
    const float* __restrict__ feats,
    const int* __restrict__ labels,
    const float* __restrict__ transitions,
    float* __restrict__ gold_out)
{
    const int wave = threadIdx.x >> 5;
    const int lane = threadIdx.x & 31;
    const int b = blockIdx.x * 8 + wave;
    const float* fb = feats + (size_t)b * (TDIM * NUM_LABELS);
    const int* lb = labels + (size_t)b * TDIM;
    float acc = 0.0f;
    for (int t = 1 + lane; t < TDIM; t += 32) {
        const int cur = lb[t], prev = lb[t - 1];
        acc += transitions[cur * NUM_LABELS + prev] + fb[t * NUM_LABELS + cur];
    }
    #pragma unroll
    for (int off = 16; off > 0; off >>= 1) acc += __shfl_xor(acc, off, 32);
    if (lane == 0) gold_out[b] = acc;
}

__global__ __launch_bounds__(256) void crf_reduce_kernel(
    const float* __restrict__ fwd,
    const float* __restrict__ gold,
    float* __restrict__ out)
{
    __shared__ float sm[256];
    float acc = 0.0f;
    for (int i = threadIdx.x; i < BDIM; i += 256) acc += fwd[i] - gold[i];
    sm[threadIdx.x] = acc;
    __syncthreads();
    #pragma unroll
    for (int s = 128; s > 0; s >>= 1) {
        if ((int)threadIdx.x < s) sm[threadIdx.x] += sm[threadIdx.x + s];
        __syncthreads();
    }
    if (threadIdx.x == 0) out[0] = sm[0] / (float)BDIM;
}

extern "C" void kernel_launch(void* const* d_in, const int* in_sizes, int n_in,
                              void* d_out, int out_size, void* d_ws, size_t ws_size,
                              hipStream_t stream) {
    const float* feats       = (const float*)d_in[0];
    const int*   label_ids   = (const int*)d_in[1];
    const float* transitions = (const float*)d_in[2];
    float* out = (float*)d_out;

    float* ws_fwd  = (float*)d_ws;          // BDIM floats
    float* ws_gold = ws_fwd + BDIM;         // BDIM floats

    // Forward scan: 32 blocks x 4 waves, each wave owns 16 batch columns.
    crf_forward_kernel<<<BDIM / (4 * 16), 128, 0, stream>>>(feats, transitions, ws_fwd);
    // Gold score: one wave per sequence.
    crf_gold_kernel<<<BDIM / 8, 256, 0, stream>>>(feats, label_ids, transitions, ws_gold);
    // Deterministic fixed-order mean.
    crf_reduce_kernel<<<1, 256, 0, stream>>>(ws_fwd, ws_gold, out);
}